// NASGNN_59330678226985
// MI455X (gfx1250) — compile-verified
//
#include <hip/hip_runtime.h>
#include <hip/hip_bf16.h>

typedef __attribute__((ext_vector_type(2))) float v2f;
typedef __attribute__((ext_vector_type(8))) float v8f;

#define DFEAT 64
#define WSTRIDE 68   // padded LDS stride (words) for transposed W: bank = 4*r + kb, conflict-free

// ---------------- degree / dinv ----------------
__global__ void deg_init_kernel(float* deg, int N) {
    int i = blockIdx.x * blockDim.x + threadIdx.x;
    if (i < N) deg[i] = 1.0f;   // self-loop contributes 1
}

__global__ void deg_accum_kernel(const int* __restrict__ dst, int E, float* deg) {
    int i = blockIdx.x * blockDim.x + threadIdx.x;
    if (i < E) atomicAdd(&deg[dst[i]], 1.0f);
}

__global__ void deg_finalize_kernel(float* deg, int N) {
    int i = blockIdx.x * blockDim.x + threadIdx.x;
    if (i < N) {
        float d = deg[i];
        d = d < 1.0f ? 1.0f : d;
        deg[i] = rsqrtf(d);     // store dinv in place
    }
}

// ---------------- fused (sum of relu'd inputs) @ W via f32 WMMA ----------------
// One wave computes a 16-row x 64-col strip of H = S @ W.
// NSRC==0: S = X0 (raw).  NSRC>=1: S = relu(P0)+...+relu(P_{NSRC-1}).
// W is staged TRANSPOSED in LDS so each B fragment is one aligned ds_load_b64.
template<int NSRC>
__global__ __launch_bounds__(256) void gcn_gemm64_wmma(
    const float* __restrict__ X0,
    const float* __restrict__ P0,
    const float* __restrict__ P1,
    const float* __restrict__ P2,
    const float* __restrict__ W,   // 64x64 row-major
    float* __restrict__ H,         // Nx64
    int N)
{
    __shared__ float Wt[DFEAT * WSTRIDE];   // Wt[col*WSTRIDE + k] = W[k][col]
    for (int i = threadIdx.x; i < DFEAT * DFEAT; i += blockDim.x) {
        int krow = i >> 6;
        int col  = i & 63;
        Wt[col * WSTRIDE + krow] = W[i];
    }
    __syncthreads();

    int wave = blockIdx.x * (blockDim.x >> 5) + (threadIdx.x >> 5);
    int lane = threadIdx.x & 31;
    int row0 = wave * 16;
    if (row0 >= N) return;                 // wave-uniform: EXEC stays all-ones

    int r    = lane & 15;                  // row within tile (A), col within tile (B/D)
    int kh   = (lane >> 4) << 1;           // 0 for lanes 0-15, 2 for lanes 16-31
    int row  = row0 + r; if (row >= N) row = N - 1;
    size_t rbase = (size_t)row * DFEAT + (size_t)kh;

    v8f acc[4] = {v8f{}, v8f{}, v8f{}, v8f{}};

    #pragma unroll
    for (int k = 0; k < DFEAT; k += 4) {
        // ---- A fragment: S[row][k+kh .. k+kh+1] (fused relu-sum, b64 loads) ----
        v2f a;
        if (NSRC == 0) {
            a = *(const v2f*)(X0 + rbase + k);
        } else {
            v2f t0 = *(const v2f*)(P0 + rbase + k);
            a.x = fmaxf(t0.x, 0.0f);
            a.y = fmaxf(t0.y, 0.0f);
            if (NSRC > 1) {
                v2f t1 = *(const v2f*)(P1 + rbase + k);
                a.x += fmaxf(t1.x, 0.0f);
                a.y += fmaxf(t1.y, 0.0f);
            }
            if (NSRC > 2) {
                v2f t2 = *(const v2f*)(P2 + rbase + k);
                a.x += fmaxf(t2.x, 0.0f);
                a.y += fmaxf(t2.y, 0.0f);
            }
        }

        // ---- B fragments: contiguous pair Wt[col][kb], Wt[col][kb+1] -> ds_load_b64 ----
        int kb = k + kh;
        #pragma unroll
        for (int t = 0; t < 4; ++t) {
            int col = t * 16 + r;
            v2f bfrag = *(const v2f*)(&Wt[col * WSTRIDE + kb]);
            acc[t] = __builtin_amdgcn_wmma_f32_16x16x4_f32(
                false, a, false, bfrag, (short)0, acc[t], false, false);
        }
    }

    // ---- store D: VGPR m -> row (row0 + mbase + m), col = t*16 + r ----
    int mbase = (lane >> 4) * 8;
    #pragma unroll
    for (int m = 0; m < 8; ++m) {
        int orow = row0 + mbase + m;
        if (orow < N) {
            size_t base = (size_t)orow * DFEAT;
            H[base +  0 + r] = acc[0][m];
            H[base + 16 + r] = acc[1][m];
            H[base + 32 + r] = acc[2][m];
            H[base + 48 + r] = acc[3][m];
        }
    }
}

// ---------------- aggregation init: out = k*b + dinv^2 * H (self-loop term) ----------------
__global__ void agg_init_kernel(const float* __restrict__ H,
                                const float* __restrict__ dinv,
                                const float* __restrict__ bias,
                                float kcount,
                                float* __restrict__ out, int N)
{
    int idx = blockIdx.x * blockDim.x + threadIdx.x;
    if (idx < N * DFEAT) {
        int v = idx >> 6;
        int c = idx & 63;
        float di = dinv[v];
        out[idx] = kcount * bias[c] + di * di * H[idx];
    }
}

// ---------------- edge scatter: out[dst] += dinv[src]*dinv[dst] * H[src] ----------------
// One wave per edge; lane handles 2 contiguous features (one b64 gather per lane).
__global__ void edge_scatter_kernel(const int* __restrict__ esrc,
                                    const int* __restrict__ edst,
                                    int E,
                                    const float* __restrict__ dinv,
                                    const float* __restrict__ H,
                                    float* __restrict__ out)
{
    int e    = (blockIdx.x * blockDim.x + threadIdx.x) >> 5;
    int lane = threadIdx.x & 31;
    if (e >= E) return;
    int s = esrc[e];
    int d = edst[e];
    float nrm = dinv[s] * dinv[d];
    size_t sb = (size_t)s * DFEAT + 2 * lane;
    size_t db = (size_t)d * DFEAT + 2 * lane;
    v2f h = *(const v2f*)(H + sb);
    atomicAdd(&out[db],     nrm * h.x);
    atomicAdd(&out[db + 1], nrm * h.y);
}

// ---------------- head: log_softmax(out3) @ W_head + b_head, gathered by label idx ----------------
__global__ void head_kernel(const float* __restrict__ out3,
                            const int* __restrict__ lbl,
                            int n_label,
                            const float* __restrict__ Wh,  // 64 x C row-major
                            const float* __restrict__ bh,
                            int C,
                            float* __restrict__ pred)
{
    int rrow = (blockIdx.x * blockDim.x + threadIdx.x) >> 5;
    int lane = threadIdx.x & 31;
    if (rrow >= n_label) return;
    int v = lbl[rrow];
    size_t base = (size_t)v * DFEAT;
    float x0 = out3[base + lane];
    float x1 = out3[base + lane + 32];

    float m = fmaxf(x0, x1);
    #pragma unroll
    for (int off = 16; off > 0; off >>= 1)
        m = fmaxf(m, __shfl_xor(m, off, 32));

    float s = __expf(x0 - m) + __expf(x1 - m);
    #pragma unroll
    for (int off = 16; off > 0; off >>= 1)
        s += __shfl_xor(s, off, 32);

    float lse = m + __logf(s);
    float l0 = x0 - lse;
    float l1 = x1 - lse;

    int c0 = lane;            // lanes 0..31, all < C when C==40
    int c1 = lane + 32;       // valid for lanes with c1 < C
    bool has0 = (c0 < C);
    bool has1 = (c1 < C);
    float acc0 = 0.0f, acc1 = 0.0f;
    #pragma unroll 4
    for (int d = 0; d < 32; ++d) {
        float va = __shfl(l0, d, 32);
        float vb = __shfl(l1, d, 32);
        if (has0) acc0 += va * Wh[d * C + c0] + vb * Wh[(d + 32) * C + c0];
        if (has1) acc1 += va * Wh[d * C + c1] + vb * Wh[(d + 32) * C + c1];
    }
    size_t ob = (size_t)rrow * C;
    if (has0) pred[ob + c0] = acc0 + bh[c0];
    if (has1) pred[ob + c1] = acc1 + bh[c1];
}

// ---------------- launch ----------------
extern "C" void kernel_launch(void* const* d_in, const int* in_sizes, int n_in,
                              void* d_out, int out_size, void* d_ws, size_t ws_size,
                              hipStream_t stream)
{
    const float* nf  = (const float*)d_in[0];
    const int*   ei  = (const int*)  d_in[1];
    const int*   lbl = (const int*)  d_in[2];
    const float* Wb[4] = { (const float*)d_in[3], (const float*)d_in[5],
                           (const float*)d_in[7], (const float*)d_in[9] };
    const float* bb[4] = { (const float*)d_in[4], (const float*)d_in[6],
                           (const float*)d_in[8], (const float*)d_in[10] };
    const float* Wh = (const float*)d_in[11];
    const float* bh = (const float*)d_in[12];

    int N  = in_sizes[0] / DFEAT;
    int E  = in_sizes[1] / 2;
    int nl = in_sizes[2];
    int C  = in_sizes[12];

    const int* esrc = ei;
    const int* edst = ei + E;

    float* ws   = (float*)d_ws;
    float* dinv = ws;
    float* H    = dinv + N;
    float* A0   = H  + (size_t)N * DFEAT;
    float* A1   = A0 + (size_t)N * DFEAT;
    float* A2   = A1 + (size_t)N * DFEAT;
    float* outb[4] = { A0, A1, A2, A0 };   // out3 reuses A0 (dead after block-3 GEMM)

    const int TB = 256;

    // degrees -> dinv
    deg_init_kernel<<<(N + TB - 1) / TB, TB, 0, stream>>>(dinv, N);
    deg_accum_kernel<<<(E + TB - 1) / TB, TB, 0, stream>>>(edst, E, dinv);
    deg_finalize_kernel<<<(N + TB - 1) / TB, TB, 0, stream>>>(dinv, N);

    int gemm_waves  = (N + 15) / 16;
    int gemm_blocks = (gemm_waves + (TB / 32) - 1) / (TB / 32);
    int elem_blocks = (int)(((size_t)N * DFEAT + TB - 1) / TB);
    int edge_blocks = (int)(((size_t)E * 32 + TB - 1) / TB);

    const float kcount[4] = { 1.0f, 1.0f, 2.0f, 3.0f };

    for (int i = 0; i < 4; ++i) {
        switch (i) {
        case 0:
            gcn_gemm64_wmma<0><<<gemm_blocks, TB, 0, stream>>>(
                nf, A0, A1, A2, Wb[0], H, N);
            break;
        case 1:
            gcn_gemm64_wmma<1><<<gemm_blocks, TB, 0, stream>>>(
                nullptr, A0, A1, A2, Wb[1], H, N);
            break;
        case 2:
            gcn_gemm64_wmma<2><<<gemm_blocks, TB, 0, stream>>>(
                nullptr, A0, A1, A2, Wb[2], H, N);
            break;
        default:
            gcn_gemm64_wmma<3><<<gemm_blocks, TB, 0, stream>>>(
                nullptr, A0, A1, A2, Wb[3], H, N);
            break;
        }
        agg_init_kernel<<<elem_blocks, TB, 0, stream>>>(
            H, dinv, bb[i], kcount[i], outb[i], N);
        edge_scatter_kernel<<<edge_blocks, TB, 0, stream>>>(
            esrc, edst, E, dinv, H, outb[i]);
    }

    int head_blocks = (int)(((size_t)nl * 32 + TB - 1) / TB);
    head_kernel<<<head_blocks, TB, 0, stream>>>(
        outb[3], lbl, nl, Wh, bh, C, (float*)d_out);
}